// MaskLanguageModeling_79663053406608
// MI455X (gfx1250) — compile-verified
//
#include <hip/hip_runtime.h>
#include <hip/hip_bf16.h>

typedef __attribute__((ext_vector_type(2))) float        v2f;
typedef __attribute__((ext_vector_type(8))) float        v8f;
typedef __attribute__((ext_vector_type(4))) unsigned int v4u;
typedef __attribute__((ext_vector_type(8))) int          v8i;
typedef __attribute__((ext_vector_type(4))) int          v4i;

#define LN_EPS 1e-5f

constexpr int D     = 1024;       // hidden dim
constexpr int S     = 512;        // sequence length
constexpr int MTOK  = 77;         // masked tokens per batch
constexpr int NROWS = 32 * MTOK;  // 2464
constexpr int V     = 32000;      // vocab
constexpr int LDA   = 1028;       // k1 LDS row stride (floats): 1028%64=4 -> conflict-free

// kernel-2 tiling
constexpr int ROWS_WG = 64;       // rows per workgroup (4 WMMA row-tiles)
constexpr int COLS_WG = 256;      // cols per workgroup (8 waves x 2 col-tiles)
constexpr int KC      = 256;      // K stage width staged via TDM
constexpr int LDB     = 260;      // TDM-padded LDS row stride (260%64=4, conflict-free)
constexpr int NSTAGE  = D / KC;   // 4
constexpr int BUFELTS = ROWS_WG * LDB;   // 16640 floats per stage buffer

// ---------------------------------------------------------------------------
// TDM 2-D load: h_ws tile [rows x KC] (row pitch D) -> LDS with pad 4 dwords
// every 256 dwords (LDS pitch = 260 floats). Descriptor per ISA 8.3/8.4.
// Rows beyond `tensor_rows` are hardware OOB -> zero-filled.
// ---------------------------------------------------------------------------
__device__ __forceinline__ void tdm_load_tile(const float* gptr,
                                              unsigned lds_byte_addr,
                                              unsigned tensor_rows) {
  const unsigned long long ga = (unsigned long long)(uintptr_t)gptr;
  v4u g0;
  g0.x = 1u;                                           // count=1, user descriptor
  g0.y = lds_byte_addr;                                // lds_addr
  g0.z = (unsigned)(ga & 0xFFFFFFFFu);                 // global_addr[31:0]
  g0.w = (unsigned)((ga >> 32) & 0x1FFFFFFu)           // global_addr[56:32]
       | (2u << 30);                                   // type = 2 ("image")
  v8i g1;
  g1[0] = (int)((2u << 16)      // data_size = 4 bytes
              | (1u << 20)      // pad_enable
              | (7u << 22)      // pad_interval: every 256 dwords
              | (3u << 25));    // pad_amount: 4 dwords  -> pitch 260 floats
  g1[1] = (int)(((unsigned)KC & 0xFFFFu) << 16);                 // tensor_dim0 lo
  g1[2] = (int)((((unsigned)KC >> 16) & 0xFFFFu)
              | ((tensor_rows & 0xFFFFu) << 16));                // dim0 hi | dim1 lo
  g1[3] = (int)(((tensor_rows >> 16) & 0xFFFFu)
              | ((unsigned)KC << 16));                           // dim1 hi | tile_dim0
  g1[4] = (int)(unsigned)ROWS_WG;                                // tile_dim1 | tile_dim2=0
  g1[5] = (int)(unsigned)D;                                      // tensor_dim0_stride lo
  g1[6] = 0;                                                     // stride0 hi | stride1 lo
  g1[7] = 0;
  v4i z4 = {};
  v8i z8 = {};
  // 6-arg form (clang-23 / therock-10.0 headers): groups 2,3 + extra group unused for 2-D
  __builtin_amdgcn_tensor_load_to_lds(g0, g1, z4, z4, z8, 0);
}

// ---------------------------------------------------------------------------
// Kernel 1: gather + GEMM1 (x@W1 + b1) + ReLU + LayerNorm -> h_ws [2464,1024]
// Grid: 154 blocks of 256 threads (8 waves). Each block: 16 rows x 1024 cols.
// ---------------------------------------------------------------------------
__global__ __launch_bounds__(256) void mlm_gemm1_ln(
    const float* __restrict__ X, const int* __restrict__ mpos,
    const float* __restrict__ W1, const float* __restrict__ b1,
    const float* __restrict__ gamma, const float* __restrict__ beta,
    float* __restrict__ h_ws)
{
  extern __shared__ float lds[];        // 16 x LDA floats (65792 B)
  const int tid  = threadIdx.x;
  const int row0 = blockIdx.x * 16;

  // ---- gather 16 token rows of X into LDS (coalesced float4) ----
  {
    const int i  = tid >> 4;
    const int cg = tid & 15;
    const int gr = row0 + i;
    const int b  = gr / MTOK;
    const int m  = gr - b * MTOK;
    const int pos = mpos[b * MTOK + m];
    const float* src = X + (size_t)(b * S + pos) * D;
#pragma unroll
    for (int j = 0; j < 16; ++j) {
      const int col = cg * 4 + j * 64;
      *(float4*)&lds[i * LDA + col] = *(const float4*)&src[col];
    }
  }
  __syncthreads();

  const int wave  = tid >> 5;
  const int lane  = tid & 31;
  const int mrow  = lane & 15;
  const int koff  = (lane >> 4) << 1;   // 0 or 2
  const int nbase = wave * 128;

  v8f acc[8] = {};

#pragma unroll 2
  for (int k0 = 0; k0 < D; k0 += 4) {
    v2f a = *(const v2f*)&lds[mrow * LDA + k0 + koff];
    const int kr = k0 + koff;
    const float* w1r0 = W1 + (size_t)kr * D;
    const float* w1r1 = w1r0 + D;
#pragma unroll
    for (int t = 0; t < 8; ++t) {
      const int n = nbase + t * 16 + mrow;
      v2f bf;
      bf.x = w1r0[n];
      bf.y = w1r1[n];
      acc[t] = __builtin_amdgcn_wmma_f32_16x16x4_f32(
          false, a, false, bf, (short)0, acc[t], false, false);
    }
  }
  __syncthreads();

  // ---- bias + ReLU into LDS as h[16][LDA] ----
#pragma unroll
  for (int t = 0; t < 8; ++t) {
    const int n = nbase + t * 16 + mrow;
    const float bias = b1[n];
#pragma unroll
    for (int v = 0; v < 8; ++v) {
      const int m = ((lane >> 4) << 3) + v;
      float val = acc[t][v] + bias;
      lds[m * LDA + n] = val > 0.0f ? val : 0.0f;
    }
  }
  __syncthreads();

  // ---- LayerNorm: 16 threads per row, shuffle reduction (wave32-safe) ----
  {
    const int r  = tid >> 4;
    const int cg = tid & 15;
    float sum = 0.0f, sumsq = 0.0f;
#pragma unroll
    for (int j = 0; j < 64; ++j) {
      const float v = lds[r * LDA + cg + j * 16];
      sum += v; sumsq += v * v;
    }
#pragma unroll
    for (int off = 8; off >= 1; off >>= 1) {
      sum   += __shfl_xor(sum,   off, 16);
      sumsq += __shfl_xor(sumsq, off, 16);
    }
    const float mu = sum * (1.0f / D);
    const float var = sumsq * (1.0f / D) - mu * mu;
    const float rs = rsqrtf(var + LN_EPS);
    float* dst = h_ws + (size_t)(row0 + r) * D;
#pragma unroll
    for (int j = 0; j < 64; ++j) {
      const int col = cg + j * 16;
      const float v = lds[r * LDA + col];
      dst[col] = (v - mu) * rs * gamma[col] + beta[col];
    }
  }
}

// ---------------------------------------------------------------------------
// Kernel 2: out = h @ W2 + b2 -> [2464, 32000]
// Grid: (125 col-blocks of 256, 39 row-blocks of 64), 256 threads (8 waves).
// Each wave: all 64 rows x 32 cols (4 row-tiles x 2 col-tiles = 8 acc tiles)
// so every W2 element loaded serves 64 rows (32 FLOP/byte from L2).
// A tile is double-buffered through LDS via the Tensor Data Mover.
// ---------------------------------------------------------------------------
__global__ __launch_bounds__(256) void mlm_gemm2(
    const float* __restrict__ h_ws, const float* __restrict__ W2,
    const float* __restrict__ b2, float* __restrict__ out)
{
  extern __shared__ float lds[];            // 2 x 64 x LDB floats (133120 B)
  const int tid  = threadIdx.x;
  const int row0 = blockIdx.y * ROWS_WG;
  const int col0 = blockIdx.x * COLS_WG;

  const int wave = tid >> 5;
  const int lane = tid & 31;
  const int mrow = lane & 15;
  const int koff = (lane >> 4) << 1;        // 0 or 2
  const int nb   = col0 + wave * 32;        // this wave's 32 columns

  const unsigned tensor_rows = (unsigned)(NROWS - row0);  // tail rows -> HW zero-fill
  const unsigned lds_base = (unsigned)(uintptr_t)&lds[0];

  // prime stage 0 (one TDM issue per workgroup; EXEC is ignored by TDM)
  if (wave == 0) {
    tdm_load_tile(h_ws + (size_t)row0 * D, lds_base, tensor_rows);
  }

  v8f acc[4][2] = {};

  for (int s = 0; s < NSTAGE; ++s) {
    if (wave == 0) __builtin_amdgcn_s_wait_tensorcnt(0);  // stage s landed
    __syncthreads();                                      // publish to all waves
    if (wave == 0 && s + 1 < NSTAGE) {                    // prefetch stage s+1
      tdm_load_tile(h_ws + (size_t)row0 * D + (s + 1) * KC,
                    lds_base + (unsigned)(((s + 1) & 1) * BUFELTS * 4),
                    tensor_rows);
    }
    const float* buf = &lds[(s & 1) * BUFELTS];
    const int ks = s * KC;

#pragma unroll 2
    for (int k0 = 0; k0 < KC; k0 += 4) {
      // 4 A fragments (one per 16-row tile), shared across both col-tiles
      v2f af[4];
#pragma unroll
      for (int rt = 0; rt < 4; ++rt)
        af[rt] = *(const v2f*)&buf[(rt * 16 + mrow) * LDB + k0 + koff];

      const int kr = ks + k0 + koff;
      const float* w2r0 = W2 + (size_t)kr * V;
      const float* w2r1 = w2r0 + V;
#pragma unroll
      for (int ct = 0; ct < 2; ++ct) {
        const int n = nb + ct * 16 + mrow;
        v2f bf;
        bf.x = w2r0[n];
        bf.y = w2r1[n];
#pragma unroll
        for (int rt = 0; rt < 4; ++rt) {
          acc[rt][ct] = __builtin_amdgcn_wmma_f32_16x16x4_f32(
              false, af[rt], false, bf, (short)0, acc[rt][ct], false, false);
        }
      }
    }
    __syncthreads();   // all waves done with buf[s&1] before stage s+2 reuses it
  }

  // ---- bias + predicated stores (64B-coalesced per half-wave) ----
#pragma unroll
  for (int ct = 0; ct < 2; ++ct) {
    const int n = nb + ct * 16 + mrow;
    const float bias = b2[n];
#pragma unroll
    for (int rt = 0; rt < 4; ++rt) {
#pragma unroll
      for (int v = 0; v < 8; ++v) {
        const int m  = rt * 16 + ((lane >> 4) << 3) + v;
        const int gr = row0 + m;
        if (gr < NROWS) out[(size_t)gr * V + n] = acc[rt][ct][v] + bias;
      }
    }
  }
}

// ---------------------------------------------------------------------------
extern "C" void kernel_launch(void* const* d_in, const int* in_sizes, int n_in,
                              void* d_out, int out_size, void* d_ws, size_t ws_size,
                              hipStream_t stream) {
  const float* X     = (const float*)d_in[0];
  const int*   mpos  = (const int*)  d_in[1];
  const float* W1    = (const float*)d_in[2];
  const float* b1    = (const float*)d_in[3];
  const float* gamma = (const float*)d_in[4];
  const float* beta  = (const float*)d_in[5];
  const float* W2    = (const float*)d_in[6];
  const float* b2    = (const float*)d_in[7];
  float* out  = (float*)d_out;
  float* h_ws = (float*)d_ws;   // 2464*1024 f32 = ~9.6 MB

  const size_t sm1 = (size_t)16 * LDA * sizeof(float);        // 65792 B
  mlm_gemm1_ln<<<NROWS / 16, 256, sm1, stream>>>(X, mpos, W1, b1, gamma, beta, h_ws);

  const size_t sm2 = (size_t)2 * BUFELTS * sizeof(float);     // 133120 B
  const int gy = (NROWS + ROWS_WG - 1) / ROWS_WG;             // 39
  mlm_gemm2<<<dim3(V / COLS_WG, gy), 256, sm2, stream>>>(h_ws, W2, b2, out);
}